// inter_gcn_670014899145
// MI455X (gfx1250) — compile-verified
//
#include <hip/hip_runtime.h>

// ---------------------------------------------------------------------------
// inter_gcn for MI455X (gfx1250, wave32)
//   B=8, C=512, H=W=64, NUM=48 nodes, per-node dim c=32
//   Bandwidth-bound: ~600 MB HBM traffic -> ~26us floor @ 23.3 TB/s.
//   Stage 1: global avg pool   (reads 201 MB, b128 coalesced)
//   Stage 2: cascade GCN       (tiny, LDS-resident, 1 block/batch) + writes
//            zero-padded 16-row A panels for the WMMA gate GEMM
//   Stage 3: FC gates via V_WMMA_F32_16X16X4_F32, branch-free K-loop
//            (2x global_load_b64 + wmma per step)
//   Stage 4: out = F*(1+e)     (reads 201 MB -- mostly L2 hits -- writes
//                               201 MB with non-temporal b128 stores so the
//                               output stream doesn't evict inputs from L2)
// ---------------------------------------------------------------------------

typedef float __attribute__((ext_vector_type(4))) f32x4;
typedef float __attribute__((ext_vector_type(2))) v2f;
typedef float __attribute__((ext_vector_type(8))) v8f;

#define B_   8
#define C_   512
#define HW_  4096
#define TC_  1536   // 3*C
#define K_   48     // graph nodes
#define CC_  32     // per-node feature dim (3C/48)
#define MPAD 16     // batch rows padded 8 -> 16 for WMMA

// ---------------- Stage 1: pooled[b, j] = mean_{hw} concat(Fv,Fd,Ft) -------
__global__ void pool_kernel(const float* __restrict__ Fv,
                            const float* __restrict__ Fd,
                            const float* __restrict__ Ft,
                            float* __restrict__ pooled) {
  __shared__ float red[256];
  const int bid = blockIdx.x;           // b*1536 + j
  const int b   = bid / TC_;
  const int j   = bid % TC_;
  const float* src = (j < C_) ? Fv : (j < 2 * C_) ? Fd : Ft;
  const int ch = j % C_;
  const f32x4* p = (const f32x4*)(src + (size_t)(b * C_ + ch) * HW_);
  const int tid = threadIdx.x;
  float s = 0.f;
#pragma unroll
  for (int i = 0; i < 4; ++i) {         // 4096 floats = 1024 f32x4 per row
    f32x4 v = p[tid + 256 * i];
    s += v.x + v.y + v.z + v.w;
  }
  red[tid] = s;
  __syncthreads();
  for (int off = 128; off > 0; off >>= 1) {
    if (tid < off) red[tid] += red[tid + off];
    __syncthreads();
  }
  if (tid == 0) pooled[bid] = red[0] * (1.0f / HW_);
}

// ---------------- Stage 2: two GCN layers + relu, one block per batch ------
// xt[k][c] layout == pooled row layout (pooled[b, k*32+c] = x[b, c, k]).
// Output: featp[g][m][n], m in 0..15 (rows 8..15 zero) -- WMMA A panels.
__global__ void gcn_kernel(const float* __restrict__ pooled,
                           const float* __restrict__ w1,
                           const float* __restrict__ w2,
                           float* __restrict__ featp) {
  __shared__ float xt[K_ * CC_];    // 48 x 32 (x transposed: row k, col c)
  __shared__ float adj[K_ * K_];    // 48 x 48
  __shared__ float sup[K_ * CC_];   // 48 x 32
  __shared__ float wg[CC_ * CC_];   // 32 x 32
  const int b = blockIdx.x;
  const int tid = threadIdx.x;

  for (int i = tid; i < K_ * CC_; i += 256) xt[i] = pooled[b * TC_ + i];

  for (int layer = 0; layer < 2; ++layer) {
    const float* W = layer ? w2 : w1;
    for (int i = tid; i < CC_ * CC_; i += 256) wg[i] = W[i];
    __syncthreads();
    // adj_pre[k][l] = <xt[k], xt[l]>
    for (int i = tid; i < K_ * K_; i += 256) {
      const int k = i / K_, l = i % K_;
      float s = 0.f;
      for (int c = 0; c < CC_; ++c) s += xt[k * CC_ + c] * xt[l * CC_ + c];
      adj[i] = s;
    }
    __syncthreads();
    // row softmax (48 rows, threads 0..47)
    if (tid < K_) {
      float m = -1e30f;
      for (int l = 0; l < K_; ++l) m = fmaxf(m, adj[tid * K_ + l]);
      float sum = 0.f;
      for (int l = 0; l < K_; ++l) {
        const float e = __expf(adj[tid * K_ + l] - m);
        adj[tid * K_ + l] = e;
        sum += e;
      }
      const float inv = 1.f / sum;
      for (int l = 0; l < K_; ++l) adj[tid * K_ + l] *= inv;
    }
    __syncthreads();
    // support = xt @ Wg  (48 x 32)
    for (int i = tid; i < K_ * CC_; i += 256) {
      const int k = i / CC_, d = i % CC_;
      float s = 0.f;
      for (int c = 0; c < CC_; ++c) s += xt[k * CC_ + c] * wg[c * CC_ + d];
      sup[i] = s;
    }
    __syncthreads();
    // xt_new = relu(adj @ support)
    for (int i = tid; i < K_ * CC_; i += 256) {
      const int k = i / CC_, d = i % CC_;
      float s = 0.f;
      for (int l = 0; l < K_; ++l) s += adj[k * K_ + l] * sup[l * CC_ + d];
      xt[i] = fmaxf(s, 0.f);
    }
    __syncthreads();
  }
  // feat[b][c*48 + k] = xt[k][c]; scatter into padded A panels:
  //   featp[g*16*512 + b*512 + j] with g = i/512, j = i%512.
  // Block b also zero-fills padding row b+8 of every gate panel.
  for (int i = tid; i < TC_; i += 256) {
    const int c = i / K_, k = i % K_;
    const int g = i / C_, j = i % C_;
    featp[(size_t)g * (MPAD * C_) + b * C_ + j]        = xt[k * CC_ + c];
    featp[(size_t)g * (MPAD * C_) + (b + 8) * C_ + j]  = 0.f;
  }
}

// ---------------- Stage 3: e_g = sigmoid(feat_g @ W_g^T + b_g) via WMMA ----
// 12 blocks x 8 waves; block -> gate = blockIdx/4 (uniform); wave -> N-tile.
// A(16x4 f32) lane layout: M = lane%16, K pair = (2*(lane/16), +1) -> b64.
// B(4x16 f32) lane layout: N = lane%16, same K pair               -> b64.
__global__ void gate_kernel(const float* __restrict__ featp,
                            const float* __restrict__ fc1w, const float* __restrict__ fc1b,
                            const float* __restrict__ fc2w, const float* __restrict__ fc2b,
                            const float* __restrict__ fc3w, const float* __restrict__ fc3b,
                            float* __restrict__ e) {
  const int tid = threadIdx.x;
  const int g   = blockIdx.x >> 2;                       // gate 0..2 (uniform)
  const int ntile = ((blockIdx.x & 3) << 3) + (tid >> 5); // 0..31
  const int n0  = ntile << 4;
  const float* W    = (g == 0) ? fc1w : (g == 1) ? fc2w : fc3w;
  const float* bias = (g == 0) ? fc1b : (g == 1) ? fc2b : fc3b;

  const int lane = tid & 31;
  const int half = lane >> 4;          // 0: K pair (0,1), 1: K pair (2,3)
  const int l16  = lane & 15;
  const float* arow = featp + (size_t)g * (MPAD * C_) + (size_t)l16 * C_;
  const float* brow = W + (size_t)(n0 + l16) * C_;

  v8f acc = {};
#pragma unroll 4
  for (int k = 0; k < C_; k += 4) {
    const int kk = k + 2 * half;       // 8-byte aligned (kk even)
    const v2f a  = *(const v2f*)(arow + kk);
    const v2f bb = *(const v2f*)(brow + kk);
    acc = __builtin_amdgcn_wmma_f32_16x16x4_f32(
        /*neg_a=*/false, a, /*neg_b=*/false, bb,
        /*c_mod=*/(short)0, acc, /*reuse_a=*/false, /*reuse_b=*/false);
  }
  // D layout: lanes 0..15, VGPR m -> D[M=m][N=l16]; covers batches 0..7.
  if (half == 0) {
    const float bv = bias[n0 + l16];
#pragma unroll
    for (int m = 0; m < B_; ++m) {
      const float v = acc[m] + bv;
      e[g * (B_ * C_) + m * C_ + (n0 + l16)] = 1.f / (1.f + __expf(-v));
    }
  }
}

// ---------------- Stage 4: out = F * (1 + e), NT stores --------------------
__global__ void scale_kernel(const float* __restrict__ Fv,
                             const float* __restrict__ Fd,
                             const float* __restrict__ Ft,
                             const float* __restrict__ e,
                             float* __restrict__ out) {
  const int bid = blockIdx.x;               // t*(B*C) + r
  const int t = bid / (B_ * C_);
  const int r = bid % (B_ * C_);
  const float* src = (t == 0) ? Fv : (t == 1) ? Fd : Ft;
  const float scale = 1.0f + e[t * (B_ * C_) + r];
  const f32x4* p = (const f32x4*)(src + (size_t)r * HW_);
  f32x4* q = (f32x4*)(out + (size_t)t * (B_ * C_) * HW_ + (size_t)r * HW_);
  const int tid = threadIdx.x;
#pragma unroll
  for (int i = 0; i < 4; ++i) {
    f32x4 v = p[tid + 256 * i];
    v *= scale;
    __builtin_nontemporal_store(v, q + tid + 256 * i);  // keep inputs in L2
  }
}

// ---------------------------------------------------------------------------
extern "C" void kernel_launch(void* const* d_in, const int* in_sizes, int n_in,
                              void* d_out, int out_size, void* d_ws, size_t ws_size,
                              hipStream_t stream) {
  (void)in_sizes; (void)n_in; (void)out_size; (void)ws_size;
  const float* Fv   = (const float*)d_in[0];
  const float* Fd   = (const float*)d_in[1];
  const float* Ft   = (const float*)d_in[2];
  const float* w1   = (const float*)d_in[3];
  const float* w2   = (const float*)d_in[4];
  const float* fc1w = (const float*)d_in[5];
  const float* fc1b = (const float*)d_in[6];
  const float* fc2w = (const float*)d_in[7];
  const float* fc2b = (const float*)d_in[8];
  const float* fc3w = (const float*)d_in[9];
  const float* fc3b = (const float*)d_in[10];

  float* ws = (float*)d_ws;
  float* pooled = ws;            // 12288 floats
  float* featp  = ws + 12288;    // 3*16*512 = 24576 floats (zero-padded A)
  float* e      = ws + 36864;    // 12288 floats
  float* out    = (float*)d_out;

  pool_kernel <<<B_ * TC_,     256, 0, stream>>>(Fv, Fd, Ft, pooled);
  gcn_kernel  <<<B_,           256, 0, stream>>>(pooled, w1, w2, featp);
  gate_kernel <<<12,           256, 0, stream>>>(featp, fc1w, fc1b, fc2w, fc2b,
                                                 fc3w, fc3b, e);
  scale_kernel<<<3 * B_ * C_,  256, 0, stream>>>(Fv, Fd, Ft, e, out);
}